// FALCONModule_9775345565959
// MI455X (gfx1250) — compile-verified
//
#include <hip/hip_runtime.h>
#include <hip/hip_bf16.h>
#include <math.h>

typedef __attribute__((ext_vector_type(2))) float v2f;
typedef __attribute__((ext_vector_type(8))) float v8f;

#define NCLASSES 8192
#define NENT     4096
#define NREL     256
#define DIM      128
#define ANON     4
#define NALL     (NENT + ANON)      // 4100
#define BATCH    8
#define NTILES   ((NALL + 15) / 16) // 257
#define NPAD     (NTILES * 16)      // 4112 (scratch stride)
#define EPSL     1e-10f

__device__ __forceinline__ float sigmoidf(float x) {
    return 1.0f / (1.0f + __expf(-x));
}

// Kernel 1: [ew1, ew2] = e_all(4100x128) @ [w1 w2](128x2), via f32 WMMA.
// One wave per 16-row tile; W padded to 16 columns. Padding correctness:
// D[m][n] depends only on A row m and B column n, so garbage pad rows/cols
// (clamped loads, aliased w columns) never reach memory — stores are guarded.
__global__ __launch_bounds__(256) void falcon_ew_wmma(
    const float* __restrict__ e_table,   // (4096,128)
    const float* __restrict__ anon_e,    // (4,128)
    const float* __restrict__ fc_w,      // (256,) = [w1 | w2]
    float* __restrict__ ew1,             // (NPAD,)
    float* __restrict__ ew2)             // (NPAD,)
{
    __shared__ float w_lds[2 * DIM];
    const int tid = threadIdx.x;
    w_lds[tid] = fc_w[tid];              // blockDim == 256 == 2*DIM
    __syncthreads();

    const int wave = tid >> 5;
    const int lane = tid & 31;
    const int half = lane >> 4;          // K-half selector
    const int mn   = lane & 15;          // A: row-in-tile (M); B: column (N)
    const int tile = blockIdx.x * 8 + wave;

    const int row  = tile * 16 + mn;
    const int lrow = (row < NALL) ? row : (NALL - 1);           // safe clamp
    const float* __restrict__ src = (lrow < NENT)
        ? (e_table + (size_t)lrow * DIM)
        : (anon_e + (size_t)(lrow - NENT) * DIM);

    // B fragment source: columns 0 (w1), 1 (w2); cols >=2 alias w1 (never stored).
    const int wbase = ((mn < 2) ? mn : 0) * DIM;

    v8f acc = {0.f, 0.f, 0.f, 0.f, 0.f, 0.f, 0.f, 0.f};

    #pragma unroll 4
    for (int k0 = 0; k0 < DIM; k0 += 4) {
        const int k = k0 + half * 2;     // this lane's K pair
        const v2f a  = *(const v2f*)(src + k);          // 8B, 8B-aligned
        const v2f bf = *(const v2f*)(&w_lds[wbase + k]);
        // D = A(16x4 f32) * B(4x16 f32) + C ; exact f32 accumulation
        acc = __builtin_amdgcn_wmma_f32_16x16x4_f32(
            /*neg_a=*/false, a, /*neg_b=*/false, bf,
            /*c_mod=*/(short)0, acc, /*reuse_a=*/false, /*reuse_b=*/false);
    }

    // D layout: lane (N=mn), VGPR v -> M = v + 8*half. Column 0 = ew1, 1 = ew2.
    if (mn < 2) {
        float* __restrict__ dst = (mn == 0) ? ew1 : ew2;
        const int base = tile * 16 + 8 * half;
        #pragma unroll
        for (int v = 0; v < 8; ++v) {
            const int r = base + v;
            if (r < NALL) dst[r] = acc[v];
        }
    }
}

// Kernel 2: single block. ci_max, per-batch scalar dots, per-batch max of
// sigmoid(ew1[j]+rw1[b]+bias+ci_max)*sigmoid(ew2[j]+cw1[b]+bias), then loss.
__global__ __launch_bounds__(256) void falcon_loss(
    const float* __restrict__ ew1,
    const float* __restrict__ ew2,
    const int*   __restrict__ x,        // (8,2) int32
    const float* __restrict__ c_table,  // (8192,128)
    const float* __restrict__ r_table,  // (256,128)
    const float* __restrict__ fc_w,     // (256,)
    const float* __restrict__ fc_b,     // (1,)
    float* __restrict__ out)            // (1,)
{
    __shared__ float red[256];
    __shared__ float s_rw1[BATCH];
    __shared__ float s_cw1[BATCH];
    __shared__ float s_cimax;

    const int tid = threadIdx.x;

    // ---- Phase 1: ci_max = max_j ew2[j] ----
    float m = -INFINITY;
    for (int j = tid; j < NALL; j += 256) m = fmaxf(m, ew2[j]);
    red[tid] = m;
    __syncthreads();
    for (int s = 128; s > 0; s >>= 1) {
        if (tid < s) red[tid] = fmaxf(red[tid], red[tid + s]);
        __syncthreads();
    }
    if (tid == 0) s_cimax = red[0];

    // ---- Phase 2: rw1[b] = r_emb[b].w1 ; cw1[b] = c_emb[b].w1 (wave b) ----
    const int wave = tid >> 5, lane = tid & 31;
    {
        const int b  = wave;                 // 8 waves, one batch row each
        const int ri = x[2 * b + 0];
        const int ci = x[2 * b + 1];
        float s1 = 0.f, s2 = 0.f;
        for (int k = lane; k < DIM; k += 32) {
            const float wk = fc_w[k];
            s1 += r_table[(size_t)ri * DIM + k] * wk;
            s2 += c_table[(size_t)ci * DIM + k] * wk;
        }
        #pragma unroll
        for (int off = 16; off > 0; off >>= 1) {
            s1 += __shfl_xor(s1, off);
            s2 += __shfl_xor(s2, off);
        }
        if (lane == 0) { s_rw1[b] = s1; s_cw1[b] = s2; }
    }
    __syncthreads();

    // ---- Phase 3: per-batch max over j, then loss ----
    const float bias  = fc_b[0];
    const float cimax = s_cimax;
    float mx[BATCH];
    #pragma unroll
    for (int b = 0; b < BATCH; ++b) mx[b] = -INFINITY;

    for (int j = tid; j < NALL; j += 256) {
        const float t1 = ew1[j] + bias + cimax;  // + rw1[b]
        const float t2 = ew2[j] + bias;          // + cw1[b]
        #pragma unroll
        for (int b = 0; b < BATCH; ++b) {
            const float v = sigmoidf(t1 + s_rw1[b]) * sigmoidf(t2 + s_cw1[b]);
            mx[b] = fmaxf(mx[b], v);
        }
    }

    float loss = 0.f;
    for (int b = 0; b < BATCH; ++b) {
        red[tid] = mx[b];
        __syncthreads();
        for (int s = 128; s > 0; s >>= 1) {
            if (tid < s) red[tid] = fmaxf(red[tid], red[tid + s]);
            __syncthreads();
        }
        if (tid == 0) loss += -logf(1.0f - red[0] + EPSL);
        __syncthreads();
    }
    if (tid == 0) out[0] = loss / (float)BATCH;
}

extern "C" void kernel_launch(void* const* d_in, const int* in_sizes, int n_in,
                              void* d_out, int out_size, void* d_ws, size_t ws_size,
                              hipStream_t stream) {
    // setup_inputs order: x, anon_e_emb, c_table, r_table, e_table, fc_w, fc_b
    const int*   x        = (const int*)  d_in[0];
    const float* anon_e   = (const float*)d_in[1];
    const float* c_table  = (const float*)d_in[2];
    const float* r_table  = (const float*)d_in[3];
    const float* e_table  = (const float*)d_in[4];
    const float* fc_w     = (const float*)d_in[5];
    const float* fc_b     = (const float*)d_in[6];
    float*       out      = (float*)d_out;

    float* ew1 = (float*)d_ws;          // NPAD floats
    float* ew2 = ew1 + NPAD;            // NPAD floats

    // 257 tiles, 8 waves (tiles) per 256-thread block -> 33 blocks
    const int nblocks = (NTILES + 7) / 8;
    falcon_ew_wmma<<<nblocks, 256, 0, stream>>>(e_table, anon_e, fc_w, ew1, ew2);
    falcon_loss<<<1, 256, 0, stream>>>(ew1, ew2, x, c_table, r_table, fc_w, fc_b, out);
}